// GCN_60129542533
// MI455X (gfx1250) — compile-verified
//
#include <hip/hip_runtime.h>
#include <hip/hip_bf16.h>

// ---------------------------------------------------------------------------
// GCN inference for MI455X (gfx1250, wave32).
// GEMMs use full-precision f32 WMMA (v_wmma_f32_16x16x4_f32): the GEMM FLOPs
// (~20 GFLOP) are negligible next to the L2-resident edge gather/scatter
// (feature matrices are 102 MB < 192 MB L2), so we keep fp32 end-to-end.
// ---------------------------------------------------------------------------

typedef __attribute__((ext_vector_type(2))) float v2f;
typedef __attribute__((ext_vector_type(8))) float v8f;

#define NODES 100000
#define HDIM  256
#define GRAPHS 64

// ---------------- generic utility kernels ----------------

__global__ void k_zero_f32(float* __restrict__ p, long long n) {
    long long i = (long long)blockIdx.x * blockDim.x + threadIdx.x;
    if (i < n) p[i] = 0.0f;
}

__global__ void k_fill_f32(float* __restrict__ p, float v, long long n) {
    long long i = (long long)blockIdx.x * blockDim.x + threadIdx.x;
    if (i < n) p[i] = v;
}

// ---------------- degree / normalization ----------------

__global__ void k_deg_count(const int* __restrict__ dst, float* __restrict__ deg, int ne) {
    long long e = (long long)blockIdx.x * blockDim.x + threadIdx.x;
    if (e < ne) atomicAdd(&deg[dst[e]], 1.0f);
}

__global__ void k_dinv(const float* __restrict__ deg, float* __restrict__ dinv, int n) {
    int i = blockIdx.x * blockDim.x + threadIdx.x;
    if (i < n) {
        float d = deg[i];                       // >= 1 because of self-loop
        dinv[i] = (d > 0.0f) ? rsqrtf(d) : 0.0f;
    }
}

// ---------------- f32 WMMA GEMM: C[M,N] = A[M,K] @ W[K,N] ----------------
// One wave -> 16(M) x 64(N) output tile (4 accumulators share the A fragment).
// Requires: M%16==0, N%64==0, K%4==0. EXEC all-ones per WMMA restriction.
//
// A fragment (16x4 f32, ISA 7.12.2): lane L holds row M=L%16;
//   lanes 0-15  -> K = k+0, k+1   (VGPR0, VGPR1)
//   lanes 16-31 -> K = k+2, k+3
// B fragment (4x16 f32): lane L holds col N=L%16;
//   lanes 0-15  -> K = k+0, k+1 ; lanes 16-31 -> K = k+2, k+3
// C/D (16x16 f32, 8 VGPRs): VGPR r = row r (lanes 0-15) / row 8+r (lanes 16-31).
__global__ __launch_bounds__(256) void k_gemm_wmma_f32(
    const float* __restrict__ A, const float* __restrict__ W,
    float* __restrict__ C, int M, int N, int K)
{
    const int lane  = threadIdx.x & 31;
    const int wave  = threadIdx.x >> 5;
    const int gwave = blockIdx.x * 8 + wave;
    const int nt64  = N >> 6;
    const int mtile = gwave / nt64;
    const int ntile = gwave % nt64;
    if (mtile * 16 >= M) return;                 // wave-uniform: EXEC stays all-ones

    const int  mrow  = mtile * 16 + (lane & 15);
    const int  khalf = (lane >> 4) << 1;         // 0 or 2
    const int  ncol  = ntile * 64 + (lane & 15);
    const float* Ap  = A + (size_t)mrow * K + khalf;

    v8f acc0 = {}, acc1 = {}, acc2 = {}, acc3 = {};

    for (int k = 0; k < K; k += 4) {
        v2f a = *(const v2f*)(Ap + k);           // A[m][k+khalf], A[m][k+khalf+1]
        const float* Wp = W + (size_t)(k + khalf) * N + ncol;
        v2f b0, b1, b2, b3;
        b0.x = Wp[0];   b0.y = Wp[N];
        b1.x = Wp[16];  b1.y = Wp[N + 16];
        b2.x = Wp[32];  b2.y = Wp[N + 32];
        b3.x = Wp[48];  b3.y = Wp[N + 48];
        acc0 = __builtin_amdgcn_wmma_f32_16x16x4_f32(false, a, false, b0, (short)0, acc0, false, false);
        acc1 = __builtin_amdgcn_wmma_f32_16x16x4_f32(false, a, false, b1, (short)0, acc1, false, false);
        acc2 = __builtin_amdgcn_wmma_f32_16x16x4_f32(false, a, false, b2, (short)0, acc2, false, false);
        acc3 = __builtin_amdgcn_wmma_f32_16x16x4_f32(false, a, false, b3, (short)0, acc3, false, false);
    }

    const int rbase = mtile * 16 + ((lane >> 4) << 3);   // row 0..7 or 8..15
    const int cbase = ntile * 64 + (lane & 15);
    for (int r = 0; r < 8; ++r) {
        size_t ro = (size_t)(rbase + r) * N;
        C[ro + cbase     ] = acc0[r];
        C[ro + cbase + 16] = acc1[r];
        C[ro + cbase + 32] = acc2[r];
        C[ro + cbase + 48] = acc3[r];
    }
}

// ---------------- edge scatter (gather h[row]*norm, atomic-add into agg[col]) ----
// One 256-thread block per edge (H == 256). Self-loops folded in: e >= ne -> node.
__global__ __launch_bounds__(HDIM) void k_scatter(
    const float* __restrict__ h, const int* __restrict__ src, const int* __restrict__ dst,
    const float* __restrict__ dinv, float* __restrict__ agg, int ne)
{
    long long e = blockIdx.x;
    int f = threadIdx.x;
    int r, c;
    if (e < ne) { r = src[e]; c = dst[e]; }
    else        { r = c = (int)(e - ne); }
    float nrm = dinv[r] * dinv[c];
    float v = h[(size_t)r * HDIM + f] * nrm;
    atomicAdd(&agg[(size_t)c * HDIM + f], v);
}

__global__ void k_bias_relu(float* __restrict__ x, const float* __restrict__ b, long long n) {
    long long i = (long long)blockIdx.x * blockDim.x + threadIdx.x;
    if (i < n) {
        float v = x[i] + b[i & (HDIM - 1)];
        x[i] = v > 0.0f ? v : 0.0f;
    }
}

// ---------------- global mean pool ----------------

__global__ void k_pool(const float* __restrict__ a, const int* __restrict__ batch,
                       float* __restrict__ sums, float* __restrict__ cnts, int n) {
    long long idx = (long long)blockIdx.x * blockDim.x + threadIdx.x;
    if (idx >= (long long)n * HDIM) return;
    int i = (int)(idx >> 8);
    int f = (int)(idx & (HDIM - 1));
    int g = batch[i];
    atomicAdd(&sums[g * HDIM + f], a[idx]);
    if (f == 0) atomicAdd(&cnts[g], 1.0f);
}

__global__ void k_head(const float* __restrict__ sums, const float* __restrict__ cnts,
                       const float* __restrict__ Wl, const float* __restrict__ bl,
                       float* __restrict__ out) {
    int t = threadIdx.x;
    if (t >= GRAPHS * 2) return;
    int g = t >> 1, c = t & 1;
    float cnt = cnts[g];
    if (cnt < 1.0f) cnt = 1.0f;
    float acc = 0.0f;
    for (int f = 0; f < HDIM; ++f)
        acc += (sums[g * HDIM + f] / cnt) * Wl[f * 2 + c];
    out[t] = acc + bl[c];
}

// ---------------- host-side orchestration ----------------

static inline long long cdiv(long long a, long long b) { return (a + b - 1) / b; }

extern "C" void kernel_launch(void* const* d_in, const int* in_sizes, int n_in,
                              void* d_out, int out_size, void* d_ws, size_t ws_size,
                              hipStream_t stream) {
    const float* x   = (const float*)d_in[0];
    const int*   ei  = (const int*)d_in[1];   // [2, E] flat; harness delivers int32
    const int*   bat = (const int*)d_in[2];
    const float* W1  = (const float*)d_in[3];
    const float* b1  = (const float*)d_in[4];
    const float* W2  = (const float*)d_in[5];
    const float* b2  = (const float*)d_in[6];
    const float* Wl  = (const float*)d_in[7];
    const float* bl  = (const float*)d_in[8];
    float* out = (float*)d_out;

    const int E  = in_sizes[1] / 2;
    const int N  = NODES;
    const int H  = HDIM;
    const int*   row = ei;
    const int*   col = ei + E;

    // workspace layout (256B-aligned slices)
    char* ws = (char*)d_ws;
    size_t off = 0;
    auto alloc = [&](size_t bytes) -> float* {
        float* p = (float*)(ws + off);
        off += (bytes + 255) & ~(size_t)255;
        return p;
    };
    float* deg  = alloc((size_t)N * sizeof(float));
    float* dinv = alloc((size_t)N * sizeof(float));
    float* bufH = alloc((size_t)N * H * sizeof(float));  // h1, then h2
    float* bufA = alloc((size_t)N * H * sizeof(float));  // agg1 -> a1
    float* bufB = alloc((size_t)N * H * sizeof(float));  // agg2 -> a2
    float* sums = alloc((size_t)GRAPHS * H * sizeof(float));
    float* cnts = alloc((size_t)GRAPHS * sizeof(float));
    (void)ws_size; (void)n_in; (void)out_size;

    const long long NH = (long long)N * H;
    const int TB = 256;

    // 1) degree with self-loops: deg = 1 + count(col)
    k_fill_f32<<<cdiv(N, TB), TB, 0, stream>>>(deg, 1.0f, N);
    k_deg_count<<<cdiv(E, TB), TB, 0, stream>>>(col, deg, E);
    k_dinv<<<cdiv(N, TB), TB, 0, stream>>>(deg, dinv, N);

    // 2) layer 1: h1 = x @ W1   (M=100000, N=256, K=128)
    {
        long long waves = ((long long)N / 16) * (H / 64);
        k_gemm_wmma_f32<<<cdiv(waves, 8), 256, 0, stream>>>(x, W1, bufH, N, H, 128);
    }
    k_zero_f32<<<cdiv(NH, TB), TB, 0, stream>>>(bufA, NH);
    k_scatter<<<E + N, H, 0, stream>>>(bufH, row, col, dinv, bufA, E);
    k_bias_relu<<<cdiv(NH, TB), TB, 0, stream>>>(bufA, b1, NH);

    // 3) layer 2: h2 = a1 @ W2  (K=256), reuse bufH
    {
        long long waves = ((long long)N / 16) * (H / 64);
        k_gemm_wmma_f32<<<cdiv(waves, 8), 256, 0, stream>>>(bufA, W2, bufH, N, H, 256);
    }
    k_zero_f32<<<cdiv(NH, TB), TB, 0, stream>>>(bufB, NH);
    k_scatter<<<E + N, H, 0, stream>>>(bufH, row, col, dinv, bufB, E);
    k_bias_relu<<<cdiv(NH, TB), TB, 0, stream>>>(bufB, b2, NH);

    // 4) global mean pool + linear head
    k_zero_f32<<<cdiv(GRAPHS * H, TB), TB, 0, stream>>>(sums, GRAPHS * H);
    k_zero_f32<<<1, TB, 0, stream>>>(cnts, GRAPHS);
    k_pool<<<cdiv(NH, TB), TB, 0, stream>>>(bufB, bat, sums, cnts, N);
    k_head<<<1, 128, 0, stream>>>(sums, cnts, Wl, bl, out);
}